// KernelMethod_SoftMax_67473936220763
// MI455X (gfx1250) — compile-verified
//
#include <hip/hip_runtime.h>
#include <hip/hip_bf16.h>

typedef __attribute__((ext_vector_type(2))) float v2f;
typedef __attribute__((ext_vector_type(8))) float v8f;

#define M_CENT 1024
#define N_PTS  131072
#define LDS_STRIDE 36   // 36 dwords/row: conflict-free ds_load_b128 across 16 cols

// ---------------------------------------------------------------------------
// Prep: per-centroid precompute.
//   B[m][0..15]  = 1/cov_d
//   B[m][16..31] = -2 * c_d / cov_d
//   c2[m] = 0.5*log2(sum_d cov_d) - 8*log2(2*pi) - 0.5*log2e * sum_d c_d^2/cov_d
// Then K[n][m] = exp2( -0.5*log2e * (A . B[m]) + c2[m] ), A = [x^2, x].
// ---------------------------------------------------------------------------
__global__ void gauss_prep(const float* __restrict__ centroids,
                           const float* __restrict__ widths,
                           const float* __restrict__ rc,
                           const float* __restrict__ rs,
                           float* __restrict__ gB,
                           float* __restrict__ gC2) {
  int m = blockIdx.x * blockDim.x + threadIdx.x;
  if (m >= M_CENT) return;
  float csq = 0.0f, covsum = 0.0f;
#pragma unroll
  for (int d = 0; d < 16; ++d) {
    float c  = centroids[m * 16 + d];
    float w  = widths[m * 16 + d];
    float cr = c * rs[d];
    float cov = w * w + rc[d] * rc[d] + cr * cr;
    float iv = 1.0f / cov;
    gB[m * 32 + d]      = iv;
    gB[m * 32 + 16 + d] = -2.0f * c * iv;
    csq    += c * c * iv;
    covsum += cov;
  }
  // 8*log2(2*pi) = 21.2119690899f ; 0.5*log2(e) = 0.7213475204f
  gC2[m] = 0.5f * log2f(covsum) - 21.2119690899f - 0.7213475204f * csq;
}

// ---------------------------------------------------------------------------
// Main fused kernel: one wave per 16-row x-tile; loop over 64 column tiles.
// WMMA f32 16x16x4, K=32 in 8 steps. exp + Z/weighted sums kept in-register.
// ---------------------------------------------------------------------------
__global__ void __launch_bounds__(256)
gauss_main(const float* __restrict__ x,
           const float* __restrict__ gB,
           const float* __restrict__ gC2,
           const float* __restrict__ coeffs,
           float* __restrict__ out) {
  extern __shared__ float smem[];
  float* sB  = smem;                               // [M][LDS_STRIDE]
  float* sC2 = smem + M_CENT * LDS_STRIDE;         // [M]
  float* sCf = sC2 + M_CENT;                       // [M]

  // ---- stage B (128KB) + per-centroid scalars into LDS ----
  const float4* gB4 = (const float4*)gB;
  for (int i = threadIdx.x; i < M_CENT * 8; i += 256) {
    int m = i >> 3, q = i & 7;
    float4 v = gB4[i];
    *(float4*)&sB[m * LDS_STRIDE + q * 4] = v;
  }
  for (int i = threadIdx.x; i < M_CENT; i += 256) {
    sC2[i] = gC2[i];
    sCf[i] = coeffs[i];
  }
  __syncthreads();

  const int wave   = threadIdx.x >> 5;
  const int lane   = threadIdx.x & 31;
  const int col16  = lane & 15;    // column within 16-col tile / x-row within tile
  const int khalf  = lane >> 4;    // which K-half this lane feeds (WMMA layout)
  const int rowBase = (blockIdx.x * 8 + wave) * 16;

  // ---- load this lane's x row (16 floats) and build A fragments ----
  const float4* xr4 = (const float4*)(x + (size_t)(rowBase + col16) * 16);
  float4 x0 = xr4[0], x1 = xr4[1], x2 = xr4[2], x3 = xr4[3];
  float xr[16] = {x0.x, x0.y, x0.z, x0.w, x1.x, x1.y, x1.z, x1.w,
                  x2.x, x2.y, x2.z, x2.w, x3.x, x3.y, x3.z, x3.w};
  // K-slot permutation: step s, lane-half h, slot j -> logical value 16h+2s+j.
  // h=0 lanes carry x^2 terms, h=1 lanes carry x terms.
  v2f afrag[8];
#pragma unroll
  for (int s = 0; s < 8; ++s) {
    float u = xr[2 * s], v = xr[2 * s + 1];
    afrag[s][0] = khalf ? u : u * u;
    afrag[s][1] = khalf ? v : v * v;
  }

  float zacc[8], wacc[8];
#pragma unroll
  for (int i = 0; i < 8; ++i) { zacc[i] = 0.0f; wacc[i] = 0.0f; }

  // ---- loop over 64 column tiles of 16 centroids ----
  for (int ct = 0; ct < M_CENT / 16; ++ct) {
    int col = ct * 16 + col16;
    const float* bp = &sB[col * LDS_STRIDE + khalf * 16];
    float4 b0 = *(const float4*)(bp);
    float4 b1 = *(const float4*)(bp + 4);
    float4 b2 = *(const float4*)(bp + 8);
    float4 b3 = *(const float4*)(bp + 12);
    float bl[16] = {b0.x, b0.y, b0.z, b0.w, b1.x, b1.y, b1.z, b1.w,
                    b2.x, b2.y, b2.z, b2.w, b3.x, b3.y, b3.z, b3.w};

    v8f acc = {};
#pragma unroll
    for (int s = 0; s < 8; ++s) {
      v2f bv;
      bv[0] = bl[2 * s];
      bv[1] = bl[2 * s + 1];
      acc = __builtin_amdgcn_wmma_f32_16x16x4_f32(
          /*neg_a=*/false, afrag[s], /*neg_b=*/false, bv,
          /*c_mod=*/(short)0, acc, /*reuse_a=*/false, /*reuse_b=*/false);
    }

    float c2v = sC2[col];
    float cfv = sCf[col];
#pragma unroll
    for (int i = 0; i < 8; ++i) {
      // K = exp2( -0.5*log2e * arg_base + c2 )
      float e = __builtin_amdgcn_exp2f(fmaf(acc[i], -0.7213475204f, c2v));
      zacc[i] += e;
      wacc[i] += e * cfv;
    }
  }

  // ---- reduce across the 16 lanes sharing each row block ----
#pragma unroll
  for (int m = 1; m < 16; m <<= 1) {
#pragma unroll
    for (int i = 0; i < 8; ++i) {
      zacc[i] += __shfl_xor(zacc[i], m, 32);
      wacc[i] += __shfl_xor(wacc[i], m, 32);
    }
  }

  // C layout: lanes 0-15 hold rows 0..7, lanes 16-31 hold rows 8..15
  if (col16 == 0) {
    int base = rowBase + khalf * 8;
#pragma unroll
    for (int i = 0; i < 8; ++i) out[base + i] = wacc[i] / zacc[i];
  }
}

extern "C" void kernel_launch(void* const* d_in, const int* in_sizes, int n_in,
                              void* d_out, int out_size, void* d_ws, size_t ws_size,
                              hipStream_t stream) {
  const float* x         = (const float*)d_in[0];
  const float* centroids = (const float*)d_in[1];
  const float* widths    = (const float*)d_in[2];
  const float* coeffs    = (const float*)d_in[3];
  const float* rc        = (const float*)d_in[4];
  const float* rs        = (const float*)d_in[5];
  float* out = (float*)d_out;

  float* gB  = (float*)d_ws;          // M*32 floats
  float* gC2 = gB + M_CENT * 32;      // M floats

  gauss_prep<<<(M_CENT + 255) / 256, 256, 0, stream>>>(centroids, widths, rc, rs,
                                                       gB, gC2);

  size_t lds_bytes = (size_t)(M_CENT * LDS_STRIDE + 2 * M_CENT) * sizeof(float);
  dim3 grid(N_PTS / (16 * 8));  // 8 waves (row tiles) per 256-thread block
  gauss_main<<<grid, 256, lds_bytes, stream>>>(x, gB, gC2, coeffs, out);
}